// SNRModule_6932077216118
// MI455X (gfx1250) — compile-verified
//
#include <hip/hip_runtime.h>
#include <math.h>

typedef float v2f __attribute__((ext_vector_type(2)));
typedef float v8f __attribute__((ext_vector_type(8)));

#define N_HID 256
// -ln(10000)/256
#define NEG_LOG1E4_OVER_D (-9.210340371976184f / 256.0f)

// sinusoidal positional-encoding value for column j at position pos = t+1
__device__ __forceinline__ float pe_val(int j, float pos) {
    float jf  = (float)(j & ~1);                  // half_idx*2 (even column of the pair)
    float arg = pos * expf(jf * NEG_LOG1E4_OVER_D);
    return (j & 1) ? cosf(arg) : sinf(arg);
}

// ---------------- kernel 0: zero workspace (deg_out, deg_in, h1, agg1) ----------
__global__ void k_zero(unsigned* __restrict__ p, int words) {
    int i = blockIdx.x * blockDim.x + threadIdx.x;
    if (i < words) p[i] = 0u;
}

// ---------------- kernel 1: degree counts over edges ----------------------------
__global__ void k_deg(const int* __restrict__ src, const int* __restrict__ dst,
                      unsigned* __restrict__ deg_out, unsigned* __restrict__ deg_in,
                      int E) {
    int e = blockIdx.x * blockDim.x + threadIdx.x;
    if (e < E) {
        atomicAdd(&deg_out[src[e]], 1u);
        atomicAdd(&deg_in[dst[e]], 1u);
    }
}

// ---------------- kernel 2: WMMA GEMV  h1[n] = (x[n,:]·W[:,1]) * rsqrt(out_deg) -
// One wave per 16-row tile; K=256 via 64 x V_WMMA_F32_16X16X4_F32.
// B fragment comes from a zero-padded [256][4] LDS image of W, so lanes with
// column >= 2 read zeros with no divergence (EXEC stays all-ones at the WMMA).
__global__ void __launch_bounds__(320) k_gemv_wmma(
    const float* __restrict__ input, const float* __restrict__ W,
    const float* __restrict__ pe_coff, const int* __restrict__ t,
    const unsigned* __restrict__ deg_out, float* __restrict__ h1, int n)
{
    __shared__ float s_pe[N_HID];
    __shared__ float s_w4[N_HID * 4];   // [k][c], c in 0..3; c>=2 is zero padding

    int tid = threadIdx.x;
    float pos = (float)(*t + 1);
    if (tid < N_HID) s_pe[tid] = pe_val(tid, pos);
    for (int i = tid; i < N_HID * 4; i += 320) {
        int k = i >> 2, c = i & 3;
        s_w4[i] = (c < 2) ? W[k * 2 + c] : 0.0f;
    }
    __syncthreads();

    int wave = tid >> 5, lane = tid & 31;
    int tiles = (n + 15) >> 4;
    int tile  = blockIdx.x * 10 + wave;
    if (tile >= tiles) return;                    // wave-uniform

    int row_base = tile << 4;
    int mloc = lane & 15;                         // A: row within tile; B: column N
    int kh   = (lane >> 4) << 1;                  // 0 for lanes 0-15, 2 for 16-31
    int row  = row_base + mloc;
    int rowc = (row < n) ? row : (n - 1);         // clamped load row (discarded if OOB)
    const float* xr = input + (size_t)rowc * N_HID;
    int cn = (mloc < 2) ? mloc : 2;               // B column select (>=2 reads zeros)
    float pc = pe_coff[0];

    v8f acc = {};
    #pragma unroll 4
    for (int k = 0; k < N_HID; k += 4) {
        int j = k + kh;
        v2f a, bm;
        a.x  = xr[j]     + pc * s_pe[j];
        a.y  = xr[j + 1] + pc * s_pe[j + 1];
        bm.x = s_w4[j * 4 + cn];
        bm.y = s_w4[(j + 1) * 4 + cn];
        // D = A(16x4,f32) x B(4x16,f32) + C(16x16,f32)
        acc = __builtin_amdgcn_wmma_f32_16x16x4_f32(
            false, a, false, bm, (short)0, acc, false, false);
    }

    // C/D layout: lane (l&15)==N column, VGPR r -> M=r (lanes 0-15) / M=r+8 (16-31).
    // Only column N=1 feeds the output gate.
    if (mloc == 1) {
        int mbase = row_base + ((lane >> 4) << 3);
        #pragma unroll
        for (int r = 0; r < 8; ++r) {
            int m = mbase + r;
            if (m < n) {
                unsigned dg = deg_out[m];
                float rs = rsqrtf((float)(dg ? dg : 1u));
                h1[m] = acc[r] * rs;
            }
        }
    }
}

// ---------------- kernel 3: edge scatter  agg1[dst] += h1[src] ------------------
__global__ void k_scatter(const int* __restrict__ src, const int* __restrict__ dst,
                          const float* __restrict__ h1, float* __restrict__ agg1,
                          int E) {
    int e = blockIdx.x * blockDim.x + threadIdx.x;
    if (e < E) atomicAdd(&agg1[dst[e]], h1[src[e]]);
}

// ---------------- kernel 4: out = (input + pc*pe) * sigmoid(relu(coef1)) --------
__global__ void __launch_bounds__(256) k_final(
    const float* __restrict__ input, const float* __restrict__ b,
    const float* __restrict__ pe_coff, const int* __restrict__ t,
    const unsigned* __restrict__ deg_in, const float* __restrict__ agg1,
    float* __restrict__ out, int n)
{
    __shared__ float s_pe[N_HID];
    __shared__ float s_gate[16];

    int tid = threadIdx.x;
    float pos = (float)(*t + 1);
    if (tid < N_HID) s_pe[tid] = pe_val(tid, pos);

    int row_base = blockIdx.x << 4;
    if (tid < 16) {
        int rowg = row_base + tid;
        float g = 0.5f;
        if (rowg < n) {
            unsigned dg = deg_in[rowg];
            float coef1 = agg1[rowg] * rsqrtf((float)(dg ? dg : 1u)) + b[1];
            float m = fmaxf(coef1, 0.0f);     // relu
            g = 1.0f / (1.0f + expf(-m));     // sigmoid
        }
        s_gate[tid] = g;
    }
    __syncthreads();

    float pc = pe_coff[0];
    const float4* in4  = (const float4*)input;
    float4*       out4 = (float4*)out;
    #pragma unroll
    for (int it = 0; it < 4; ++it) {
        int q    = it * 256 + tid;       // float4 index within 16x256 tile (0..1023)
        int rloc = q >> 6;               // 64 float4 per row
        int row  = row_base + rloc;
        if (row >= n) continue;
        int c4 = q & 63;
        size_t idx = (size_t)row * (N_HID / 4) + c4;
        float4 v = in4[idx];
        int j = c4 << 2;
        float g = s_gate[rloc];
        v.x = (v.x + pc * s_pe[j])     * g;
        v.y = (v.y + pc * s_pe[j + 1]) * g;
        v.z = (v.z + pc * s_pe[j + 2]) * g;
        v.w = (v.w + pc * s_pe[j + 3]) * g;
        out4[idx] = v;
    }
}

extern "C" void kernel_launch(void* const* d_in, const int* in_sizes, int n_in,
                              void* d_out, int out_size, void* d_ws, size_t ws_size,
                              hipStream_t stream)
{
    const float* input   = (const float*)d_in[0];
    const float* W       = (const float*)d_in[1];
    const float* b       = (const float*)d_in[2];
    const float* pe_coff = (const float*)d_in[3];
    const int*   esrc    = (const int*)d_in[4];
    const int*   edst    = (const int*)d_in[5];
    const int*   t       = (const int*)d_in[6];

    int n = in_sizes[0] / N_HID;     // 100000
    int E = in_sizes[4];             // 3200000

    // workspace layout: [deg_out u32[n] | deg_in u32[n] | h1 f32[n] | agg1 f32[n]]
    unsigned* deg_out = (unsigned*)d_ws;
    unsigned* deg_in  = deg_out + n;
    float*    h1      = (float*)(deg_in + n);
    float*    agg1    = h1 + n;

    int words = 4 * n;               // zero everything (h1 zeroing is harmless)
    k_zero<<<(words + 255) / 256, 256, 0, stream>>>((unsigned*)d_ws, words);

    k_deg<<<(E + 255) / 256, 256, 0, stream>>>(esrc, edst, deg_out, deg_in, E);

    int tiles   = (n + 15) / 16;                 // 6250
    int blocks2 = (tiles + 9) / 10;              // 10 waves/block -> 625
    k_gemv_wmma<<<blocks2, 320, 0, stream>>>(input, W, pe_coff, t, deg_out, h1, n);

    k_scatter<<<(E + 255) / 256, 256, 0, stream>>>(esrc, edst, h1, agg1, E);

    k_final<<<tiles, 256, 0, stream>>>(input, b, pe_coff, t, deg_in, agg1,
                                       (float*)d_out, n);
}